// FCOSInference_37409165148441
// MI455X (gfx1250) — compile-verified
//
#include <hip/hip_runtime.h>
#include <hip/hip_bf16.h>
#include <cstdint>
#include <math.h>

// ---------------- problem constants ----------------
#define NIMG   16
#define NCLS   80
#define NLVL   3
#define KSEL   1000            // PRE_NMS_TOP_N per level (HW*C >= 50000 > 1000 on all levels)
#define POST_N 100
#define M_TOT  (NLVL * KSEL)   // 3000 candidates per image
#define PRE_TH_LOGIT (-2.9444389791664403f)  // logit(0.05)
#define NMS_TH 0.6f
#define IMG_WF 800.0f
#define IMG_HF 800.0f

__constant__ int   c_HW[3]     = {10000, 2500, 625};
__constant__ float c_STRIDE[3] = {8.0f, 16.0f, 32.0f};

__device__ __forceinline__ float sigf(float x) { return 1.0f / (1.0f + expf(-x)); }

// types for CDNA5 TDM builtin
typedef __attribute__((ext_vector_type(4))) unsigned int v4u;
typedef __attribute__((ext_vector_type(4))) int          v4i;
typedef __attribute__((ext_vector_type(8))) int          v8i;

// ----------------------------------------------------------------------------
// workspace layout (d_ws):
//   uws[0..47]    : B1 (13-bit bin of float bits[31:19]) per (img,lvl)
//   uws[48..95]   : countStrictlyAbove(B1)
//   uws[96..143]  : K26 = (B1<<13)|B2  (26-bit cutoff key on bits[31:6])
//   uws[144..191] : strictTotal = count of keys strictly above K26
//   floats from +256: cand[N][6][M_TOT] planes: 0=score 1=x1 2=y1 3=x2 4=y2 5=cls
// ----------------------------------------------------------------------------

__global__ void k_init(float* cand) {
  int i = blockIdx.x * blockDim.x + threadIdx.x;
  if (i < NIMG * 6 * M_TOT) cand[i] = 0.0f;
}

// thread0 finds, scanning 8192-bin hist from the top, the first bin where the
// suffix count reaches `target`; writes bin and strictly-above count.
__device__ void find_thresh(unsigned* hist, unsigned* chunk, unsigned target,
                            unsigned* gBin, unsigned* gStrict) {
  unsigned cs = 0;
  int base = threadIdx.x * 32;
  #pragma unroll
  for (int j = 0; j < 32; ++j) cs += hist[base + j];
  chunk[threadIdx.x] = cs;
  __syncthreads();
  if (threadIdx.x == 0) {
    unsigned acc = 0; int cidx;
    for (cidx = 255; cidx >= 0; --cidx) {
      if (acc + chunk[cidx] >= target) break;
      acc += chunk[cidx];
    }
    int B = 0;
    if (cidx >= 0) {
      int b;
      for (b = cidx * 32 + 31; b >= cidx * 32; --b) {
        if (acc + hist[b] >= target) break;
        acc += hist[b];
      }
      B = (b < cidx * 32) ? cidx * 32 : b;
    }
    *gBin = (unsigned)B;
    *gStrict = acc;
  }
  __syncthreads();
}

// pass 1: 8192-bin histogram of float-bits[31:19] of candidate scores
__global__ void k_hist1(const float* __restrict__ cls0, const float* __restrict__ cls1,
                        const float* __restrict__ cls2,
                        const float* __restrict__ ctr0, const float* __restrict__ ctr1,
                        const float* __restrict__ ctr2, unsigned* uws) {
  __shared__ unsigned hist[8192];
  __shared__ unsigned chunk[256];
  int n = blockIdx.x / 3, lvl = blockIdx.x % 3;
  int HW = c_HW[lvl];
  const float* clsP[3] = {cls0, cls1, cls2};
  const float* ctrP[3] = {ctr0, ctr1, ctr2};
  const float* cls = clsP[lvl] + (size_t)n * NCLS * HW;
  const float* ctr = ctrP[lvl] + (size_t)n * HW;
  for (int i = threadIdx.x; i < 8192; i += blockDim.x) hist[i] = 0u;
  __syncthreads();
  int tot = NCLS * HW;
  for (int e = threadIdx.x; e < tot; e += blockDim.x) {
    int pe = e + 8 * 256;
    if (pe < tot) __builtin_prefetch(&cls[pe], 0, 1);  // -> global_prefetch_b8
    float cl = cls[e];
    if (cl > PRE_TH_LOGIT) {
      int hw = e % HW;
      float s = sigf(cl) * sigf(ctr[hw]);
      atomicAdd(&hist[__float_as_uint(s) >> 19], 1u);
    }
  }
  __syncthreads();
  find_thresh(hist, chunk, (unsigned)KSEL, &uws[blockIdx.x], &uws[48 + blockIdx.x]);
}

// pass 2: refine within bin B1 on bits[18:6]
__global__ void k_hist2(const float* __restrict__ cls0, const float* __restrict__ cls1,
                        const float* __restrict__ cls2,
                        const float* __restrict__ ctr0, const float* __restrict__ ctr1,
                        const float* __restrict__ ctr2, unsigned* uws) {
  __shared__ unsigned hist[8192];
  __shared__ unsigned chunk[256];
  __shared__ unsigned sB2, sAcc2;
  int n = blockIdx.x / 3, lvl = blockIdx.x % 3;
  int HW = c_HW[lvl];
  const float* clsP[3] = {cls0, cls1, cls2};
  const float* ctrP[3] = {ctr0, ctr1, ctr2};
  const float* cls = clsP[lvl] + (size_t)n * NCLS * HW;
  const float* ctr = ctrP[lvl] + (size_t)n * HW;
  unsigned B1 = uws[blockIdx.x];
  unsigned cntGT = uws[48 + blockIdx.x];
  for (int i = threadIdx.x; i < 8192; i += blockDim.x) hist[i] = 0u;
  __syncthreads();
  int tot = NCLS * HW;
  for (int e = threadIdx.x; e < tot; e += blockDim.x) {
    int pe = e + 8 * 256;
    if (pe < tot) __builtin_prefetch(&cls[pe], 0, 1);
    float cl = cls[e];
    if (cl > PRE_TH_LOGIT) {
      int hw = e % HW;
      unsigned bits = __float_as_uint(sigf(cl) * sigf(ctr[hw]));
      if ((bits >> 19) == B1) atomicAdd(&hist[(bits >> 6) & 8191u], 1u);
    }
  }
  __syncthreads();
  unsigned target = (cntGT < (unsigned)KSEL) ? ((unsigned)KSEL - cntGT) : 1u;
  find_thresh(hist, chunk, target, &sB2, &sAcc2);
  if (threadIdx.x == 0) {
    uws[96 + blockIdx.x]  = (B1 << 13) | sB2;   // 26-bit cutoff key
    uws[144 + blockIdx.x] = cntGT + sAcc2;      // strictly-above count
  }
}

// pass 3: collect selected candidates, decode boxes
__global__ void k_collect(const float* __restrict__ cls0, const float* __restrict__ cls1,
                          const float* __restrict__ cls2,
                          const float* __restrict__ ctr0, const float* __restrict__ ctr1,
                          const float* __restrict__ ctr2,
                          const float* __restrict__ reg0, const float* __restrict__ reg1,
                          const float* __restrict__ reg2,
                          const float* __restrict__ loc0, const float* __restrict__ loc1,
                          const float* __restrict__ loc2,
                          const unsigned* uws, float* cand) {
  __shared__ unsigned cntS, cntT;
  int n = blockIdx.x / 3, lvl = blockIdx.x % 3;
  int HW = c_HW[lvl];
  float stride = c_STRIDE[lvl];
  const float* clsP[3] = {cls0, cls1, cls2};
  const float* ctrP[3] = {ctr0, ctr1, ctr2};
  const float* regP[3] = {reg0, reg1, reg2};
  const float* locP[3] = {loc0, loc1, loc2};
  const float* cls = clsP[lvl] + (size_t)n * NCLS * HW;
  const float* ctr = ctrP[lvl] + (size_t)n * HW;
  const float* reg = regP[lvl] + (size_t)n * 4 * HW;
  const float* loc = locP[lvl];
  unsigned K26 = uws[96 + blockIdx.x];
  unsigned strictTotal = uws[144 + blockIdx.x];
  if (threadIdx.x == 0) { cntS = 0u; cntT = 0u; }
  __syncthreads();
  float* cp = cand + (size_t)n * 6 * M_TOT;
  int tot = NCLS * HW;
  for (int e = threadIdx.x; e < tot; e += blockDim.x) {
    int pe = e + 8 * 256;
    if (pe < tot) __builtin_prefetch(&cls[pe], 0, 1);
    float cl = cls[e];
    if (cl <= PRE_TH_LOGIT) continue;
    int c = e / HW, hw = e - c * HW;
    float val = sigf(cl) * sigf(ctr[hw]);
    unsigned key = __float_as_uint(val) >> 6;
    int slot = -1;
    if (key > K26) {
      slot = (int)atomicAdd(&cntS, 1u);
    } else if (key == K26) {
      unsigned t = atomicAdd(&cntT, 1u);
      if (strictTotal + t < (unsigned)KSEL) slot = (int)(strictTotal + t);
    }
    if (slot >= 0 && slot < KSEL) {
      float x = loc[2 * hw], y = loc[2 * hw + 1];
      float l = reg[0 * HW + hw] * stride, tt = reg[1 * HW + hw] * stride;
      float r = reg[2 * HW + hw] * stride, b  = reg[3 * HW + hw] * stride;
      float x1 = fminf(fmaxf(x - l, 0.f), IMG_WF);
      float y1 = fminf(fmaxf(y - tt, 0.f), IMG_HF);
      float x2 = fminf(fmaxf(x + r, 0.f), IMG_WF);
      float y2 = fminf(fmaxf(y + b, 0.f), IMG_HF);
      int i = lvl * KSEL + slot;
      cp[0 * M_TOT + i] = sqrtf(val);
      cp[1 * M_TOT + i] = x1;
      cp[2 * M_TOT + i] = y1;
      cp[3 * M_TOT + i] = x2;
      cp[4 * M_TOT + i] = y2;
      cp[5 * M_TOT + i] = (float)(c + 1);
    }
  }
}

// pass 4: greedy batched NMS, one workgroup per image, candidates in LDS
__global__ void k_nms(const float* __restrict__ cand, float* __restrict__ out) {
  extern __shared__ float sm[];
  float* S  = sm;
  float* X1 = sm + 1 * M_TOT;
  float* Y1 = sm + 2 * M_TOT;
  float* X2 = sm + 3 * M_TOT;
  float* Y2 = sm + 4 * M_TOT;
  float* CL = sm + 5 * M_TOT;
  float* redV  = sm + 6 * M_TOT;          // 256
  int*   redI  = (int*)(sm + 6 * M_TOT + 256);
  float* pickS = (float*)(redI + 256);    // 100
  int*   pickI = (int*)(pickS + POST_N);  // 100

  int n = blockIdx.x;
  const float* g = cand + (size_t)n * 6 * M_TOT;

#if defined(__has_builtin)
#if __has_builtin(__builtin_amdgcn_tensor_load_to_lds) && __has_builtin(__builtin_amdgcn_s_wait_tensorcnt)
  // CDNA5 Tensor Data Mover: DMA the 6*3000-float candidate block into LDS.
  // 1-row 2D tile, data_size=4B, tile_dim0 = tensor_dim0 = 18000 elements.
  if ((threadIdx.x >> 5) == 0) {  // one wave issues; TENSORcnt is per-wave
    unsigned lds_off = (unsigned)(unsigned long long)(uintptr_t)sm;
    unsigned long long ga = (unsigned long long)(uintptr_t)g;
    v4u g0 = { 1u,                                      // count=1 valid D#
               lds_off,                                 // lds_addr
               (unsigned)ga,                            // global_addr[31:0]
               ((unsigned)(ga >> 32) & 0x1FFFFFFu) | (2u << 30) }; // addr[56:32] | type=2
    v8i g1 = { (int)0x00020000u,   // workgroup_mask=0, data_size=2 (4B)
               (int)0x46500000u,   // tensor_dim0[15:0]=18000 in bits[63:48]
               (int)0x00010000u,   // tensor_dim0 hi=0 | tensor_dim1 lo=1
               (int)0x46500000u,   // tensor_dim1 hi=0 | tile_dim0=18000
               1,                  // tile_dim1=1, tile_dim2=0
               18000,              // tensor_dim0_stride lo32
               (int)0x46500000u,   // stride0 hi16=0 | stride1 lo16=18000
               0 };
    v4i z4 = {0, 0, 0, 0};
#if __clang_major__ >= 23
    v8i z8 = {0, 0, 0, 0, 0, 0, 0, 0};
    __builtin_amdgcn_tensor_load_to_lds(g0, g1, z4, z4, z8, 0);
#else
    __builtin_amdgcn_tensor_load_to_lds(g0, g1, z4, z4, 0);
#endif
    __builtin_amdgcn_s_wait_tensorcnt(0);
  }
  __syncthreads();
#endif
#endif
  // guaranteed-correct copy (overwrites TDM result with identical data)
  for (int k = threadIdx.x; k < 6 * M_TOT; k += blockDim.x) sm[k] = g[k];
  __syncthreads();

  for (int k = threadIdx.x; k < M_TOT; k += blockDim.x)
    if (!(S[k] > 0.0f)) S[k] = -1.0f;
  __syncthreads();

  for (int it = 0; it < POST_N; ++it) {
    // argmax with first-index tie-break (matches jnp.argmax)
    float bv = -3.0f; int bi = M_TOT;
    for (int k = threadIdx.x; k < M_TOT; k += blockDim.x) {
      float v = S[k];
      if (v > bv) { bv = v; bi = k; }
    }
    redV[threadIdx.x] = bv; redI[threadIdx.x] = bi;
    __syncthreads();
    for (int off = 128; off > 0; off >>= 1) {
      if ((int)threadIdx.x < off) {
        float v2 = redV[threadIdx.x + off]; int i2 = redI[threadIdx.x + off];
        float v1 = redV[threadIdx.x];       int i1 = redI[threadIdx.x];
        if (v2 > v1 || (v2 == v1 && i2 < i1)) { redV[threadIdx.x] = v2; redI[threadIdx.x] = i2; }
      }
      __syncthreads();
    }
    int p = redI[0]; float ps = redV[0];
    if (threadIdx.x == 0) { pickI[it] = p; pickS[it] = ps; }
    float bx1 = X1[p], by1 = Y1[p], bx2 = X2[p], by2 = Y2[p], bc = CL[p];
    float barea = (bx2 - bx1) * (by2 - by1);
    // suppress: class-offset NMS == same-class IoU, else IoU 0
    for (int k = threadIdx.x; k < M_TOT; k += blockDim.x) {
      if (CL[k] == bc) {
        float xx1 = fmaxf(bx1, X1[k]), yy1 = fmaxf(by1, Y1[k]);
        float xx2 = fminf(bx2, X2[k]), yy2 = fminf(by2, Y2[k]);
        float inter = fmaxf(xx2 - xx1, 0.f) * fmaxf(yy2 - yy1, 0.f);
        float ar = (X2[k] - X1[k]) * (Y2[k] - Y1[k]);
        float iou = inter / (ar + barea - inter + 1e-9f);
        if (iou > NMS_TH) S[k] = -1.0f;
      }
    }
    if (threadIdx.x == 0) S[p] = -1.0f;
    __syncthreads();
  }

  // outputs: boxes [16,100,4] | scores [16,100] | classes [16,100] | keep [16,100]
  for (int it = threadIdx.x; it < POST_N; it += blockDim.x) {
    int p = pickI[it]; float sc = pickS[it];
    bool keep = sc > 0.0f;
    float* ob = out + (size_t)n * POST_N * 4 + (size_t)it * 4;
    ob[0] = keep ? X1[p] : 0.f;
    ob[1] = keep ? Y1[p] : 0.f;
    ob[2] = keep ? X2[p] : 0.f;
    ob[3] = keep ? Y2[p] : 0.f;
    out[NIMG * POST_N * 4 + n * POST_N + it] = keep ? sc : 0.f;
    out[NIMG * POST_N * 5 + n * POST_N + it] = keep ? CL[p] : 0.f;
    out[NIMG * POST_N * 6 + n * POST_N + it] = keep ? 1.f : 0.f;
  }
}

extern "C" void kernel_launch(void* const* d_in, const int* in_sizes, int n_in,
                              void* d_out, int out_size, void* d_ws, size_t ws_size,
                              hipStream_t stream) {
  const float* loc0 = (const float*)d_in[0];
  const float* cls0 = (const float*)d_in[1];
  const float* reg0 = (const float*)d_in[2];
  const float* ctr0 = (const float*)d_in[3];
  const float* loc1 = (const float*)d_in[4];
  const float* cls1 = (const float*)d_in[5];
  const float* reg1 = (const float*)d_in[6];
  const float* ctr1 = (const float*)d_in[7];
  const float* loc2 = (const float*)d_in[8];
  const float* cls2 = (const float*)d_in[9];
  const float* reg2 = (const float*)d_in[10];
  const float* ctr2 = (const float*)d_in[11];

  unsigned* uws = (unsigned*)d_ws;
  float* cand = (float*)d_ws + 256;  // 1 KB past control words

  int initN = NIMG * 6 * M_TOT;
  k_init<<<(initN + 255) / 256, 256, 0, stream>>>(cand);
  k_hist1<<<NIMG * NLVL, 256, 0, stream>>>(cls0, cls1, cls2, ctr0, ctr1, ctr2, uws);
  k_hist2<<<NIMG * NLVL, 256, 0, stream>>>(cls0, cls1, cls2, ctr0, ctr1, ctr2, uws);
  k_collect<<<NIMG * NLVL, 256, 0, stream>>>(cls0, cls1, cls2, ctr0, ctr1, ctr2,
                                             reg0, reg1, reg2, loc0, loc1, loc2,
                                             uws, cand);
  size_t shmem = (size_t)(6 * M_TOT + 256) * 4 + 256 * 4 + POST_N * 8;
  k_nms<<<NIMG, 256, shmem, stream>>>(cand, (float*)d_out);
}